// TabularGNN_45346264711451
// MI455X (gfx1250) — compile-verified
//
#include <hip/hip_runtime.h>
#include <math.h>

#define GCN_DIM 128
#define WPITCH 144  // pair-row stride in float2s: 1152B % 256 == 128 -> half-wave bank split

typedef __attribute__((ext_vector_type(2))) float v2f;
typedef __attribute__((ext_vector_type(8))) float v8f;

// ---------------- CSR build (by destination) ----------------

__global__ void k_zero_counts(int* __restrict__ counts, int n) {
  int i = blockIdx.x * blockDim.x + threadIdx.x;
  if (i < n) counts[i] = 0;
}

__global__ void k_count(const int* __restrict__ dst, int* __restrict__ counts, int e) {
  int i = blockIdx.x * blockDim.x + threadIdx.x;
  if (i < e) {
    __hip_atomic_fetch_add(&counts[dst[i]], 1, __ATOMIC_RELAXED, __HIP_MEMORY_SCOPE_AGENT);
  }
}

// Single-workgroup exclusive scan over counts -> rowstart[N+1]; also copy to cursor.
__launch_bounds__(1024)
__global__ void k_scan_rowstart(const int* __restrict__ counts, int* __restrict__ rowstart,
                                int* __restrict__ cursor, int n) {
  __shared__ int sd[1024];
  __shared__ int s_carry;
  const int tid = threadIdx.x;
  if (tid == 0) s_carry = 0;
  __syncthreads();
  for (int base = 0; base < n; base += 1024) {
    int i = base + tid;
    int v = (i < n) ? counts[i] : 0;
    sd[tid] = v;
    __syncthreads();
    for (int off = 1; off < 1024; off <<= 1) {  // Hillis-Steele inclusive scan
      int t = (tid >= off) ? sd[tid - off] : 0;
      __syncthreads();
      sd[tid] += t;
      __syncthreads();
    }
    int incl = sd[tid];
    int excl = incl - v;
    int carry = s_carry;
    if (i < n) {
      rowstart[i] = carry + excl;
      cursor[i]   = carry + excl;
    }
    __syncthreads();
    if (tid == 1023) s_carry = carry + incl;
    __syncthreads();
  }
  if (tid == 0) rowstart[n] = s_carry;
}

// dinv[i] = rsqrt(counts[i] + 1)   (+1 = self-loop; always > 0)
__global__ void k_dinv(const int* __restrict__ counts, float* __restrict__ dinv, int n) {
  int i = blockIdx.x * blockDim.x + threadIdx.x;
  if (i < n) dinv[i] = rsqrtf((float)counts[i] + 1.0f);
}

// adjsrc[cursor[dst[e]]++] = src[e]
__global__ void k_fill(const int* __restrict__ src, const int* __restrict__ dst,
                       int* __restrict__ cursor, int* __restrict__ adjsrc, int e) {
  int i = blockIdx.x * blockDim.x + threadIdx.x;
  if (i < e) {
    int d = dst[i];
    int pos = __hip_atomic_fetch_add(&cursor[d], 1, __ATOMIC_RELAXED, __HIP_MEMORY_SCOPE_AGENT);
    adjsrc[pos] = src[i];
  }
}

// ---------------- fp32 WMMA GEMM: C[n_rows x 128] = A[n_rows x 128] @ W[128 x 128] ----------------
// Block: 256 threads = 8 waves; each wave owns a 16-row strip x all 128 cols.
// W staged in LDS pair-interleaved so each B fragment is one ds_load_b64 straight
// into an even-aligned VGPR pair (no shuffle movs). V_WMMA_F32_16X16X4_F32, 32 K-steps.

__launch_bounds__(256)
__global__ void k_gemm128_wmma(const float* __restrict__ A, const float* __restrict__ W,
                               float* __restrict__ C, int n_rows) {
  __shared__ float2 sWp[64 * WPITCH];  // 72 KB: sWp[k>>1][c] = {W[k][c], W[k+1][c]}
  {
    for (int idx = threadIdx.x; idx < GCN_DIM * GCN_DIM / 4; idx += blockDim.x) {
      int k  = idx >> 5;           // 32 float4 per 128-col row
      int c4 = (idx & 31) * 4;
      float4 w = ((const float4*)W)[idx];
      int p = k >> 1, sub = k & 1;
      float* base = (float*)&sWp[p * WPITCH + c4];
      base[0 * 2 + sub] = w.x;
      base[1 * 2 + sub] = w.y;
      base[2 * 2 + sub] = w.z;
      base[3 * 2 + sub] = w.w;
    }
  }
  __syncthreads();

  const int wave = threadIdx.x >> 5;   // 0..7
  const int lane = threadIdx.x & 31;
  const int lo   = lane & 15;          // 0..15
  const int hi   = lane >> 4;          // 0 or 1
  const int row_base = blockIdx.x * 128 + wave * 16;

  int arow = row_base + lo;
  int arow_c = arow < n_rows ? arow : (n_rows - 1);  // clamp ragged tail (safe reads)
  const float* Arow = A + (size_t)arow_c * GCN_DIM;

  v8f acc[8];
#pragma unroll
  for (int j = 0; j < 8; ++j) acc[j] = (v8f)(0.0f);

  for (int k0 = 0; k0 < GCN_DIM; k0 += 4) {
    // 16x4 f32 A layout: lane(lo)=row; VGPR0: K=k0+2*hi, VGPR1: K=k0+2*hi+1
    float2 af = *(const float2*)(Arow + k0 + 2 * hi);
    v2f a;
    a.x = af.x;
    a.y = af.y;
    const float2* brow = &sWp[((k0 >> 1) + hi) * WPITCH + lo];
#pragma unroll
    for (int j = 0; j < 8; ++j) {
      float2 bf = brow[j * 16];  // ds_load_b64, immediate offset j*128B
      v2f b;
      b.x = bf.x;
      b.y = bf.y;
      acc[j] = __builtin_amdgcn_wmma_f32_16x16x4_f32(
          false, a, false, b, (short)0, acc[j], false, false);
    }
  }

  // C/D 16x16 f32 layout: VGPR v, lanes 0-15 -> row v, lanes 16-31 -> row v+8; col = lo
#pragma unroll
  for (int j = 0; j < 8; ++j) {
    const int cb = j * 16;
#pragma unroll
    for (int v = 0; v < 8; ++v) {
      int r = row_base + v + 8 * hi;
      if (r < n_rows) C[(size_t)r * GCN_DIM + cb + lo] = acc[j][v];
    }
  }
}

// ---------------- gather aggregation (no atomics) ----------------
// One wave32 per destination node:
//   OUT[d][:] = relu( dinv[d] * ( dinv[d]*H[d][:] + sum_{s in in(d)} dinv[s]*H[s][:] ) + bias[:] )
__launch_bounds__(256)
__global__ void k_gather_agg(const int* __restrict__ rowstart, const int* __restrict__ adjsrc,
                             const float* __restrict__ dinv, const float* __restrict__ H,
                             const float* __restrict__ bias, float* __restrict__ OUT, int n) {
  int nid  = blockIdx.x * (blockDim.x >> 5) + (threadIdx.x >> 5);
  int lane = threadIdx.x & 31;
  if (nid >= n) return;

  float dd = dinv[nid];
  int rs = rowstart[nid];
  int re = rowstart[nid + 1];

  // self-loop term: dinv[d] * H[d]
  float4 h0 = ((const float4*)(H + (size_t)nid * GCN_DIM))[lane];
  float4 acc;
  acc.x = h0.x * dd; acc.y = h0.y * dd; acc.z = h0.z * dd; acc.w = h0.w * dd;

  int j = rs;
  for (; j + 2 <= re; j += 2) {  // 2-way unroll to hide row-load latency
    int s0 = adjsrc[j];
    int s1 = adjsrc[j + 1];
    float w0 = dinv[s0];
    float w1 = dinv[s1];
    float4 v0 = ((const float4*)(H + (size_t)s0 * GCN_DIM))[lane];
    float4 v1 = ((const float4*)(H + (size_t)s1 * GCN_DIM))[lane];
    acc.x += v0.x * w0 + v1.x * w1;
    acc.y += v0.y * w0 + v1.y * w1;
    acc.z += v0.z * w0 + v1.z * w1;
    acc.w += v0.w * w0 + v1.w * w1;
  }
  if (j < re) {
    int s0 = adjsrc[j];
    float w0 = dinv[s0];
    float4 v0 = ((const float4*)(H + (size_t)s0 * GCN_DIM))[lane];
    acc.x += v0.x * w0; acc.y += v0.y * w0; acc.z += v0.z * w0; acc.w += v0.w * w0;
  }

  int c4 = lane * 4;
  float4 r;
  r.x = fmaxf(acc.x * dd + bias[c4 + 0], 0.0f);
  r.y = fmaxf(acc.y * dd + bias[c4 + 1], 0.0f);
  r.z = fmaxf(acc.z * dd + bias[c4 + 2], 0.0f);
  r.w = fmaxf(acc.w * dd + bias[c4 + 3], 0.0f);
  ((float4*)(OUT + (size_t)nid * GCN_DIM))[lane] = r;
}

// out[i][c] = out[i][c] (== x@Wp) + bp[c] + R[i][c]   (R already relu(agg2+b2))
__global__ void k_final(float* __restrict__ out, const float* __restrict__ R,
                        const float* __restrict__ bp, int n) {
  int t = blockIdx.x * blockDim.x + threadIdx.x;  // over n*32 float4's
  if (t < n * 32) {
    int c4 = (t & 31) * 4;
    float4 p = ((const float4*)out)[t];
    float4 h = ((const float4*)R)[t];
    float4 r;
    r.x = p.x + bp[c4 + 0] + h.x;
    r.y = p.y + bp[c4 + 1] + h.y;
    r.z = p.z + bp[c4 + 2] + h.z;
    r.w = p.w + bp[c4 + 3] + h.w;
    ((float4*)out)[t] = r;
  }
}

// ---------------- launch ----------------

extern "C" void kernel_launch(void* const* d_in, const int* in_sizes, int n_in,
                              void* d_out, int out_size, void* d_ws, size_t ws_size,
                              hipStream_t stream) {
  const float* x  = (const float*)d_in[0];
  const int*   ei = (const int*)d_in[1];     // [2, E] flattened (JAX demotes int64->int32)
  // d_in[2] = batch (unused)
  const float* W1 = (const float*)d_in[3];
  const float* b1 = (const float*)d_in[4];
  const float* W2 = (const float*)d_in[5];
  const float* b2 = (const float*)d_in[6];
  const float* Wp = (const float*)d_in[7];
  const float* bp = (const float*)d_in[8];
  float* out = (float*)d_out;

  const int N = in_sizes[0] / GCN_DIM;       // 100000
  const int E = in_sizes[1] / 2;             // 1600000
  const int* src = ei;
  const int* dst = ei + E;

  // workspace layout (256B-aligned slabs):
  // counts[N] | rowstart[N+1] | cursor[N] | adjsrc[E] | dinv[N] | buf1[N*128] | buf2[N*128]
  char* p = (char*)d_ws;
  auto alloc = [&](size_t bytes) {
    char* r = p;
    p += (bytes + 255) & ~(size_t)255;
    return r;
  };
  int*   counts   = (int*)alloc((size_t)N * 4);
  int*   rowstart = (int*)alloc((size_t)(N + 1) * 4);
  int*   cursor   = (int*)alloc((size_t)N * 4);
  int*   adjsrc   = (int*)alloc((size_t)E * 4);
  float* dinv     = (float*)alloc((size_t)N * 4);
  float* buf1     = (float*)alloc((size_t)N * GCN_DIM * 4);
  float* buf2     = (float*)alloc((size_t)N * GCN_DIM * 4);

  const int T = 256;
  dim3 blkT(T);
  int gN    = (N + T - 1) / T;
  int gE    = (E + T - 1) / T;
  int gNd   = (N * 32 + T - 1) / T;            // N*128 floats as float4
  int gNw   = (N + (T / 32) - 1) / (T / 32);   // one wave per node
  int gGemm = (N + 127) / 128;

  // CSR build + normalization (reused by both layers)
  k_zero_counts<<<gN, blkT, 0, stream>>>(counts, N);
  k_count<<<gE, blkT, 0, stream>>>(dst, counts, E);
  k_scan_rowstart<<<1, 1024, 0, stream>>>(counts, rowstart, cursor, N);
  k_dinv<<<gN, blkT, 0, stream>>>(counts, dinv, N);
  k_fill<<<gE, blkT, 0, stream>>>(src, dst, cursor, adjsrc, E);

  // layer 1: buf1 = x@W1 ; buf2 = relu(gather(buf1) + b1)
  k_gemm128_wmma<<<gGemm, blkT, 0, stream>>>(x, W1, buf1, N);
  k_gather_agg<<<gNw, blkT, 0, stream>>>(rowstart, adjsrc, dinv, buf1, b1, buf2, N);

  // layer 2: buf1 = buf2@W2 ; buf2 = relu(gather(buf1) + b2)
  k_gemm128_wmma<<<gGemm, blkT, 0, stream>>>(buf2, W2, buf1, N);
  k_gather_agg<<<gNw, blkT, 0, stream>>>(rowstart, adjsrc, dinv, buf1, b2, buf2, N);

  // projection + fuse: out = x@Wp ; out += bp + buf2
  k_gemm128_wmma<<<gGemm, blkT, 0, stream>>>(x, Wp, out, N);
  k_final<<<gNd, blkT, 0, stream>>>(out, buf2, bp, N);
}